// KVGather_18511309046302
// MI455X (gfx1250) — compile-verified
//
#include <hip/hip_runtime.h>

// KV gather + per-tile scale:
//   out[b,i,t,:,:] = kv[b, r_idx[b,i,t], :, :] * r_weight[b,i,t]
// n=32, p2=49, topk=4, hw=64, c=256. One 64KB tile per workgroup.
// Pure bandwidth kernel: B128 loads (RT, keep kv hot in 192MB L2),
// B128 non-temporal stores (411MB write-once stream must not evict kv),
// scalar SMEM loads for the per-tile index/weight, 16 loads in flight/thread.

typedef float v4f __attribute__((ext_vector_type(4)));

constexpr int P2   = 49;
constexpr int TOPK = 4;
constexpr int HW   = 64;
constexpr int C    = 256;
constexpr int TILE_ELEMS       = HW * C;                       // 16384 floats = 64 KB
constexpr int THREADS          = 256;                          // 8 wave32 per block
constexpr int VECS_PER_THREAD  = TILE_ELEMS / 4 / THREADS;     // 16 float4 per thread

__global__ __launch_bounds__(THREADS)
void kvgather_scale_kernel(const int*   __restrict__ r_idx,
                           const float* __restrict__ r_weight,
                           const float* __restrict__ kv,
                           float*       __restrict__ out) {
  const int tile = blockIdx.x;                 // flat (b, i, t)
  const int b    = tile / (P2 * TOPK);

  // Uniform per block -> scalar loads (s_load) on gfx1250.
  const int   j = r_idx[tile];                 // selected region in [0, P2)
  const float w = r_weight[tile];

  const v4f* __restrict__ src =
      (const v4f*)(kv + ((size_t)b * P2 + (size_t)j) * TILE_ELEMS);
  v4f* __restrict__ dst = (v4f*)(out + (size_t)tile * TILE_ELEMS);

  const int t = threadIdx.x;

  // gfx1250 global_prefetch_b8: pull the tile's first lines toward L2/L0 early.
  __builtin_prefetch(&src[t], 0, 3);

  // Batch all 16 B128 loads so they are simultaneously in flight (MLP),
  // then scale and stream out with non-temporal B128 stores.
  v4f v[VECS_PER_THREAD];
#pragma unroll
  for (int i = 0; i < VECS_PER_THREAD; ++i) {
    v[i] = src[t + i * THREADS];
  }
#pragma unroll
  for (int i = 0; i < VECS_PER_THREAD; ++i) {
    v4f r = v[i] * w;
    __builtin_nontemporal_store(r, &dst[t + i * THREADS]);
  }
}

extern "C" void kernel_launch(void* const* d_in, const int* in_sizes, int n_in,
                              void* d_out, int out_size, void* d_ws, size_t ws_size,
                              hipStream_t stream) {
  const int*   r_idx    = (const int*)  d_in[0];   // (n, p2, topk) int
  const float* r_weight = (const float*)d_in[1];   // (n, p2, topk) f32
  const float* kv       = (const float*)d_in[2];   // (n, p2, hw, c) f32
  float*       out      = (float*)d_out;           // (n, p2, topk, hw, c) f32

  const int tiles = in_sizes[0];                   // n * p2 * topk = 6272
  kvgather_scale_kernel<<<tiles, THREADS, 0, stream>>>(r_idx, r_weight, kv, out);
}